// SoftAttention_11544872091692
// MI455X (gfx1250) — compile-verified
//
#include <hip/hip_runtime.h>

#define SEQ 65536
#define DIM 1024
#define TS 16
#define TSTRIDE 1028          // floats per LDS tile row; 1028 % 64 == 4 -> conflict-free WMMA A gather
#define THREADS 256           // 8 waves (wave32)
#define MAIN_BLOCKS 512       // 4096 tiles of 16 rows -> 8 tiles per block

typedef __attribute__((ext_vector_type(2))) float v2f;
typedef __attribute__((ext_vector_type(8))) float v8f;

// ---------------------------------------------------------------------------
// Zero the accumulation targets (harness poisons d_out / d_ws with 0xAA).
// ws layout: ws[0..1023] = u = W1^T q ; ws[1024] = c0
// ---------------------------------------------------------------------------
__global__ void init_kernel(float* __restrict__ ws, float* __restrict__ out) {
    int i = blockIdx.x * blockDim.x + threadIdx.x;
    if (i < DIM) out[i] = 0.0f;
    if (i < DIM + 1) ws[i] = 0.0f;
}

// ---------------------------------------------------------------------------
// u[k] = sum_d q[d] * W1[d,k]   (W1 row-major [DIM,DIM])
// grid = (4, 8): blockIdx.x picks 256 k-columns (coalesced), blockIdx.y a
// 128-row d-slab; partial sums combined with float atomics (8K atomics total).
// ---------------------------------------------------------------------------
__global__ void qW1_kernel(const float* __restrict__ W1, const float* __restrict__ q,
                           float* __restrict__ ws) {
    const int k  = blockIdx.x * THREADS + threadIdx.x;
    const int d0 = blockIdx.y * 128;
    float acc = 0.0f;
    for (int d = d0; d < d0 + 128; ++d)
        acc = fmaf(q[d], W1[(size_t)d * DIM + k], acc);
    atomicAdd(&ws[k], acc);
}

// ---------------------------------------------------------------------------
// c0 = sum_d q[d] * (sum_k W2[d,k]*vn[k]) + b * sum_d q[d]
// ---------------------------------------------------------------------------
__global__ void c0_kernel(const float* __restrict__ W2, const float* __restrict__ q,
                          const float* __restrict__ v,  const float* __restrict__ b,
                          float* __restrict__ ws) {
    __shared__ float red[THREADS];
    const float* vn = v + (size_t)(SEQ - 1) * DIM;   // last timestep
    const int tid = threadIdx.x;
    float acc = 0.0f;
    const int dbase = blockIdx.x * 64;
    for (int d = dbase; d < dbase + 64; ++d) {
        float partial = 0.0f;
        for (int kk = tid; kk < DIM; kk += THREADS)
            partial = fmaf(W2[(size_t)d * DIM + kk], vn[kk], partial);
        acc = fmaf(q[d], partial, acc);              // linear: safe to scale partials
    }
    if (blockIdx.x == 0) {                           // fold in b * sum(q) once
        const float bb = b[0];
        for (int kk = tid; kk < DIM; kk += THREADS)
            acc = fmaf(q[kk], bb, acc);
    }
    red[tid] = acc;
    __syncthreads();
    for (int s = THREADS / 2; s > 0; s >>= 1) {
        if (tid < s) red[tid] += red[tid + s];
        __syncthreads();
    }
    if (tid == 0) atomicAdd(&ws[DIM], red[0]);
}

// ---------------------------------------------------------------------------
// Async DMA one 16-row tile (16 x 1024 f32) global -> LDS, non-temporal.
// Each thread issues one b128 per row (lane-contiguous 512B per wave per row).
// Tracked by ASYNCcnt; caller must s_wait_asynccnt + barrier before use.
// ---------------------------------------------------------------------------
__device__ __forceinline__ void async_load_tile(const float* __restrict__ src,
                                                float* __restrict__ dst_lds, int tid) {
    const float* g = src + tid * 4;
    float* l = dst_lds + tid * 4;
    #pragma unroll
    for (int r = 0; r < TS; ++r) {
        unsigned long long ga = (unsigned long long)(uintptr_t)(g + r * DIM);
        unsigned int       la = (unsigned int)(uintptr_t)(l + r * TSTRIDE);
        asm volatile("global_load_async_to_lds_b128 %0, %1, off th:TH_LOAD_NT"
                     :: "v"(la), "v"(ga) : "memory");
    }
}

__device__ __forceinline__ void wait_async_all() {
    asm volatile("s_wait_asynccnt 0x0" ::: "memory");
}

// ---------------------------------------------------------------------------
// Fused streaming pass over v (single 256 MB read), double-buffered in LDS:
//   per 16-row tile: scores16 = V_tile @ u + c0   (V_WMMA_F32_16X16X4_F32,
//     B = u broadcast across all 16 columns -> every C column = scores)
//   then out_partial += scores16^T @ V_tile       (FMA from LDS)
// Each of the 8 waves owns a 128-column d-slice for both phases; tile t+1 is
// async-DMA'd into the alternate LDS buffer while tile t is computed.
// ---------------------------------------------------------------------------
__global__ void __launch_bounds__(THREADS)
main_kernel(const float* __restrict__ v, const float* __restrict__ ws,
            float* __restrict__ out) {
    __shared__ __align__(16) float tile[2][TS * TSTRIDE];   // 2 x 16 x 1028 f32
    __shared__ __align__(16) float u_lds[DIM];
    __shared__ float swave[8][16];
    __shared__ float sfin[16];

    const int tid  = threadIdx.x;
    const int wave = tid >> 5;
    const int lane = tid & 31;

    for (int i = tid; i < DIM; i += THREADS) u_lds[i] = ws[i];
    const float c0 = ws[DIM];

    float acc0 = 0.0f, acc1 = 0.0f, acc2 = 0.0f, acc3 = 0.0f;

    const int kbase = wave * 128;            // this wave's d-slice base
    const int dbase = kbase + lane;          // this lane's first owned column
    const int row   = lane & 15;             // WMMA A: lane -> matrix row M
    const int koff  = (lane >> 4) << 1;      // lanes 16..31 hold K=2,3

    // prologue: prefetch first tile into buffer 0
    async_load_tile(v + (size_t)blockIdx.x * TS * DIM, &tile[0][0], tid);

    int buf = 0;
    for (int t = blockIdx.x; t < SEQ / TS; t += MAIN_BLOCKS) {
        wait_async_all();                    // tile[buf] resident in LDS
        __syncthreads();                     // all waves loaded + done with prior compute

        const int tn = t + MAIN_BLOCKS;      // uniform across block
        if (tn < SEQ / TS)                   // prefetch next tile into other buffer
            async_load_tile(v + (size_t)tn * TS * DIM, &tile[buf ^ 1][0], tid);

        // ---- phase 1: partial scores via WMMA over this wave's 128 d's ----
        v8f c = {};
        const float* arow = &tile[buf][row * TSTRIDE + koff];   // A fragment base
        const float* brow = &u_lds[kbase + koff];               // B fragment base (u replicated)
        #pragma unroll
        for (int k = 0; k < 128; k += 4) {
            v2f a = *(const v2f*)(&arow[kbase + k]);
            v2f b = *(const v2f*)(&brow[k]);
            c = __builtin_amdgcn_wmma_f32_16x16x4_f32(
                    /*neg_a=*/false, a, /*neg_b=*/false, b,
                    /*c_mod=*/(short)0, c, /*reuse_a=*/false, /*reuse_b=*/false);
        }
        // C[m][n] identical for all n; lane 0 holds M=0..7 in c[0..7],
        // lane 16 holds M=8..15.
        if (lane == 0) {
            #pragma unroll
            for (int r = 0; r < 8; ++r) swave[wave][r] = c[r];
        } else if (lane == 16) {
            #pragma unroll
            for (int r = 0; r < 8; ++r) swave[wave][8 + r] = c[r];
        }
        __syncthreads();
        if (tid < 16) {                       // reduce 8 wave-partials + c0
            float s = c0;
            #pragma unroll
            for (int w = 0; w < 8; ++w) s += swave[w][tid];
            sfin[tid] = s;
        }
        __syncthreads();

        // ---- phase 2: out[d] += sum_m score[m] * V[m,d] (lane-coalesced LDS) ----
        #pragma unroll
        for (int m = 0; m < 16; ++m) {
            const float sm = sfin[m];
            const float* tr = &tile[buf][m * TSTRIDE + dbase];
            acc0 = fmaf(sm, tr[0],  acc0);
            acc1 = fmaf(sm, tr[32], acc1);
            acc2 = fmaf(sm, tr[64], acc2);
            acc3 = fmaf(sm, tr[96], acc3);
        }
        buf ^= 1;
    }

    atomicAdd(&out[dbase],      acc0);
    atomicAdd(&out[dbase + 32], acc1);
    atomicAdd(&out[dbase + 64], acc2);
    atomicAdd(&out[dbase + 96], acc3);
}

// ---------------------------------------------------------------------------
extern "C" void kernel_launch(void* const* d_in, const int* in_sizes, int n_in,
                              void* d_out, int out_size, void* d_ws, size_t ws_size,
                              hipStream_t stream) {
    (void)in_sizes; (void)n_in; (void)out_size; (void)ws_size;
    const float* v  = (const float*)d_in[0];
    const float* W1 = (const float*)d_in[1];
    const float* W2 = (const float*)d_in[2];
    const float* q  = (const float*)d_in[3];
    const float* b  = (const float*)d_in[4];
    float* out = (float*)d_out;
    float* ws  = (float*)d_ws;

    init_kernel<<<dim3(5),            dim3(256),     0, stream>>>(ws, out);
    qW1_kernel <<<dim3(4, 8),         dim3(THREADS), 0, stream>>>(W1, q, ws);
    c0_kernel  <<<dim3(16),           dim3(THREADS), 0, stream>>>(W2, q, v, b, ws);
    main_kernel<<<dim3(MAIN_BLOCKS),  dim3(THREADS), 0, stream>>>(v, ws, out);
}